// GCN_layer_83872121357058
// MI455X (gfx1250) — compile-verified
//
#include <hip/hip_runtime.h>
#include <math.h>

typedef float v8f __attribute__((ext_vector_type(8)));
typedef float v2f __attribute__((ext_vector_type(2)));

#define D_FEAT 128

// ---------------------------------------------------------------------------
// K0: zero the accumulator (d_out) and the degree buffer (d_ws).
// ---------------------------------------------------------------------------
__global__ void gcn_zero_kernel(float* __restrict__ out, float* __restrict__ deg,
                                int n_out4, int n_deg) {
    int i = blockIdx.x * blockDim.x + threadIdx.x;
    int stride = gridDim.x * blockDim.x;
    float4 z = make_float4(0.f, 0.f, 0.f, 0.f);
    float4* out4 = (float4*)out;
    for (int j = i; j < n_out4; j += stride) out4[j] = z;
    for (int j = i; j < n_deg; j += stride) deg[j] = 0.f;
}

// ---------------------------------------------------------------------------
// K1: deg[tgt] += ew[e]   (global_atomic_add_f32, no return)
// ---------------------------------------------------------------------------
__global__ void gcn_deg_kernel(const int* __restrict__ edge,
                               const float* __restrict__ ew,
                               float* __restrict__ deg, int E) {
    int e = blockIdx.x * blockDim.x + threadIdx.x;
    if (e < E) {
        int tgt = edge[2 * E + e];
        unsafeAtomicAdd(&deg[tgt], ew[e]);
    }
}

// ---------------------------------------------------------------------------
// K2: wave-per-edge SpMM scatter.
//   lane l gathers x[src][4l..4l+3] (global_load_b128, L2-resident) and does
//   4 coalesced global_atomic_add_f32 into out[tgt][4l..4l+3].
// ---------------------------------------------------------------------------
__global__ __launch_bounds__(256) void gcn_spmm_kernel(
    const float* __restrict__ x, const int* __restrict__ edge,
    const float* __restrict__ ew, const float* __restrict__ deg,
    float* __restrict__ out, int E) {
    int gtid = blockIdx.x * blockDim.x + threadIdx.x;
    int e    = gtid >> 5;
    int lane = gtid & 31;
    if (e >= E) return;

    int   src = edge[e];             // row 0: source
    int   tgt = edge[2 * E + e];     // row 2: target
    float w   = ew[e] / deg[tgt];    // deg[tgt] > 0 whenever this edge exists

    const float4* xp = (const float4*)(x + (size_t)src * D_FEAT) + lane;
    float4 v = *xp;

    float* op = out + (size_t)tgt * D_FEAT + lane * 4;
    unsafeAtomicAdd(op + 0, v.x * w);
    unsafeAtomicAdd(op + 1, v.y * w);
    unsafeAtomicAdd(op + 2, v.z * w);
    unsafeAtomicAdd(op + 3, v.w * w);
}

// ---------------------------------------------------------------------------
// K3: ReLU + row L2-normalize, 16 rows per wave via WMMA Gram trick.
//
// A (16x4 f32) operand layout: lane L (half = L/16, m = L%16) holds
//   VGPR0 = row(base+m) feat(4k + 2*half), VGPR1 = feat(4k + 2*half + 1).
// B (4x16) of A-transposed has the identical per-lane register image, so we
// pass the same v2f as both operands: C += A*A^T accumulates the Gram matrix;
// diag(C) = squared L2 norms of the 16 ReLU'd rows.
// ---------------------------------------------------------------------------
__global__ __launch_bounds__(32) void gcn_norm_kernel(float* __restrict__ out, int n) {
    int base = blockIdx.x * 16;
    int lane = threadIdx.x;
    int half = lane >> 4;
    int m    = lane & 15;

    int rowA = base + m;
    if (rowA >= n) rowA = n - 1;             // safe clamp (unused when 16 | n)
    const float* rowp = out + (size_t)rowA * D_FEAT;

    v8f c = {};
    #pragma unroll
    for (int kc = 0; kc < D_FEAT / 4; ++kc) {
        int f = kc * 4 + half * 2;
        v2f a;
        a.x = fmaxf(rowp[f + 0], 0.f);       // ReLU before the norm, per reference
        a.y = fmaxf(rowp[f + 1], 0.f);
        // c = a * a^T + c   -> emits v_wmma_f32_16x16x4_f32
        c = __builtin_amdgcn_wmma_f32_16x16x4_f32(
            /*neg_a=*/false, a, /*neg_b=*/false, a,
            /*c_mod=*/(short)0, c, /*reuse_a=*/false, /*reuse_b=*/false);
    }

    // Extract diag(C): row r<8 -> c[r] @ lane r ; row r>=8 -> c[r-8] @ lane r+16.
    float inv[16];
    #pragma unroll
    for (int r = 0; r < 16; ++r) {
        int srcLane = (r < 8) ? r : (r + 16);
        float s = __shfl(c[r & 7], srcLane, 32);
        inv[r] = 1.0f / fmaxf(sqrtf(s), 1e-12f);   // F.normalize eps semantics
    }

    // In-place: out = relu(out) * inv_norm. Lane covers feats [4*lane, 4*lane+3].
    #pragma unroll
    for (int r = 0; r < 16; ++r) {
        int row = base + r;
        if (row >= n) break;
        float4* p = (float4*)(out + (size_t)row * D_FEAT) + lane;
        float4 v = *p;
        v.x = fmaxf(v.x, 0.f) * inv[r];
        v.y = fmaxf(v.y, 0.f) * inv[r];
        v.z = fmaxf(v.z, 0.f) * inv[r];
        v.w = fmaxf(v.w, 0.f) * inv[r];
        *p = v;
    }
}

// ---------------------------------------------------------------------------
extern "C" void kernel_launch(void* const* d_in, const int* in_sizes, int n_in,
                              void* d_out, int out_size, void* d_ws, size_t ws_size,
                              hipStream_t stream) {
    const float* x    = (const float*)d_in[0];   // [N, 128] f32
    const int*   edge = (const int*)d_in[1];     // [3, E]   int32
    const float* ew   = (const float*)d_in[2];   // [E]      f32
    float*       out  = (float*)d_out;           // [N, 128] f32 (accum + final)
    float*       deg  = (float*)d_ws;            // [N]      f32 scratch

    const int N = in_sizes[0] / D_FEAT;
    const int E = in_sizes[2];

    // K0: zero accumulator + degrees
    {
        int n_out4 = N * (D_FEAT / 4);
        gcn_zero_kernel<<<1024, 256, 0, stream>>>(out, deg, n_out4, N);
    }
    // K1: degree scatter-add
    gcn_deg_kernel<<<(E + 255) / 256, 256, 0, stream>>>(edge, ew, deg, E);
    // K2: wave-per-edge scatter SpMM
    {
        long long threads = (long long)E * 32;
        int blocks = (int)((threads + 255) / 256);
        gcn_spmm_kernel<<<blocks, 256, 0, stream>>>(x, edge, ew, deg, out, E);
    }
    // K3: WMMA-based ReLU + L2 row normalize (16 rows per wave)
    gcn_norm_kernel<<<(N + 15) / 16, 32, 0, stream>>>(out, N);
}